// NearestNeighborGraph_2473901162619
// MI455X (gfx1250) — compile-verified
//
#include <hip/hip_runtime.h>
#include <float.h>

typedef float v2f __attribute__((ext_vector_type(2)));
typedef float v8f __attribute__((ext_vector_type(8)));

#define NSAMP 16
#define NP    2048
#define DIMS  64
#define KNN   16
#define TILE  16
#define NTILE (NP / TILE)   // 128 j-tiles (and row-tiles) per batch

// ---------------------------------------------------------------------------
// Kernel 1: squared norms  x2[b*NP + p] = ||h[b,p,:]||^2
// ---------------------------------------------------------------------------
__global__ __launch_bounds__(256) void knn_norms_kernel(
    const float* __restrict__ h, float* __restrict__ x2) {
  int p = blockIdx.x * blockDim.x + threadIdx.x;
  if (p < NSAMP * NP) {
    const float* v = h + (size_t)p * DIMS;
    float s = 0.0f;
#pragma unroll
    for (int d = 0; d < DIMS; ++d) s += v[d] * v[d];
    x2[p] = s;
  }
}

// ---------------------------------------------------------------------------
// Kernel 2: per (batch, row-tile) workgroup.
//   Phase A: 8 waves compute 16x2048 distance block via chained
//            V_WMMA_F32_16X16X4_F32 (K=64 -> 16 chained WMMAs per tile),
//            results staged in 128KB of LDS (CDNA5 320KB WGP LDS).
//   Phase B: each wave extracts K=16 smallest (with index, tie -> lower j)
//            for 2 of the 16 rows.
// ---------------------------------------------------------------------------
__global__ __launch_bounds__(256) void knn_main_kernel(
    const float* __restrict__ h, const float* __restrict__ x2,
    float* __restrict__ knn_dist, int* __restrict__ dst_out,
    int* __restrict__ src_out) {
  extern __shared__ float sdist[];   // [TILE][NP] = 16*2048 floats = 128KB

  const int b    = blockIdx.x >> 7;        // /128
  const int it   = blockIdx.x & (NTILE - 1);
  const int i0   = it * TILE;
  const int tid  = threadIdx.x;
  const int wave = tid >> 5;               // 0..7
  const int lane = tid & 31;
  const int lhalf = lane >> 4;             // 0/1 : selects K pair {0,1} vs {2,3}
  const int lmod  = lane & 15;             // row (A) / col (B) within tile

  const float* hb  = h  + (size_t)b * NP * DIMS;
  const float* x2b = x2 + b * NP;

  // --- A fragments: 16 K-steps of the 16-row i-tile, loaded once per wave.
  // 32-bit A 16x4 layout: lanes 0-15 hold M=lane, K=4s+{0,1};
  //                       lanes 16-31 hold M=lane-16, K=4s+{2,3}.
  v2f afrag[16];
  {
    const float* arow = hb + (size_t)(i0 + lmod) * DIMS + 2 * lhalf;
#pragma unroll
    for (int s = 0; s < 16; ++s) {
      afrag[s].x = arow[4 * s + 0];
      afrag[s].y = arow[4 * s + 1];
    }
  }
  // x2 of the i-rows this lane's C fragment touches: M = r + 8*lhalf
  float x2i[8];
#pragma unroll
  for (int r = 0; r < 8; ++r) x2i[r] = x2b[i0 + r + 8 * lhalf];

  // --- Phase A: sweep j-tiles (wave w handles tiles w, w+8, ...)
#pragma unroll 1
  for (int jt = wave; jt < NTILE; jt += 8) {
    const int j0 = jt * TILE;
    // B 4x16 layout mirrors A: lane n holds column N=lmod, K=4s+2*lhalf+{0,1}
    const float* brow = hb + (size_t)(j0 + lmod) * DIMS + 2 * lhalf;

    v8f c = {0.f, 0.f, 0.f, 0.f, 0.f, 0.f, 0.f, 0.f};
#pragma unroll
    for (int s = 0; s < 16; ++s) {
      v2f bf;
      bf.x = brow[4 * s + 0];
      bf.y = brow[4 * s + 1];
      // D = A*B + C  (fp32, 16x16x4), accumulator chained over K
      c = __builtin_amdgcn_wmma_f32_16x16x4_f32(
          /*neg_a=*/false, afrag[s], /*neg_b=*/false, bf,
          /*c_mod=*/(short)0, c, /*reuse_a=*/false, /*reuse_b=*/false);
    }

    const float x2j = x2b[j0 + lmod];
    // C/D layout: VGPR r, lanes<16 -> M=r ; lanes>=16 -> M=r+8 ; N=lmod
#pragma unroll
    for (int r = 0; r < 8; ++r) {
      const int m = r + 8 * lhalf;
      const float dval = x2i[r] + x2j - 2.0f * c[r];
      sdist[m * NP + j0 + lmod] = dval;
    }
  }
  __syncthreads();

  // --- Phase B: top-K extraction. Wave w owns rows w and w+8.
#pragma unroll 1
  for (int rr = 0; rr < 2; ++rr) {
    const int m  = wave + rr * 8;
    const int gi = i0 + m;                         // row index within batch
    float* row = sdist + m * NP;
    const size_t outbase = ((size_t)b * NP + gi) * KNN;

#pragma unroll 1
    for (int k = 0; k < KNN; ++k) {
      // per-lane min over strided columns
      float best = FLT_MAX;
      int   bidx = 0x7FFFFFFF;
#pragma unroll 1
      for (int j = lane; j < NP; j += 32) {
        const float v = row[j];
        if (v < best) { best = v; bidx = j; }
      }
      // wave32 butterfly min-with-index, tie -> smaller index
#pragma unroll
      for (int off = 16; off > 0; off >>= 1) {
        const float ov = __shfl_xor(best, off, 32);
        const int   oi = __shfl_xor(bidx, off, 32);
        if (ov < best || (ov == best && oi < bidx)) { best = ov; bidx = oi; }
      }
      if (lane == 0) {
        knn_dist[outbase + k] = best;
        dst_out[outbase + k]  = bidx + b * NP;
        src_out[outbase + k]  = gi + b * NP;
        row[bidx] = FLT_MAX;   // exclude from subsequent rounds
      }
      __syncthreads();         // uniform across all waves: 2*16 barriers each
    }
  }
}

// ---------------------------------------------------------------------------
extern "C" void kernel_launch(void* const* d_in, const int* in_sizes, int n_in,
                              void* d_out, int out_size, void* d_ws, size_t ws_size,
                              hipStream_t stream) {
  const float* h = (const float*)d_in[0];
  // d_in[1] is K (== 16), fixed at compile time here.

  float* x2 = (float*)d_ws;                 // 16*2048 floats = 128KB scratch

  const int S = NSAMP * NP * KNN;           // 524288 elems per output
  float* knn_dist = (float*)d_out;
  int*   dst_out  = (int*)d_out + S;
  int*   src_out  = (int*)d_out + 2 * S;

  {
    const int n = NSAMP * NP;
    knn_norms_kernel<<<(n + 255) / 256, 256, 0, stream>>>(h, x2);
  }
  {
    const size_t shmem = (size_t)TILE * NP * sizeof(float);  // 131072 B
    knn_main_kernel<<<NSAMP * NTILE, 256, shmem, stream>>>(
        h, x2, knn_dist, dst_out, src_out);
  }
}